// GcnBlock_53206054863486
// MI455X (gfx1250) — compile-verified
//
#include <hip/hip_runtime.h>
#include <hip/hip_bf16.h>

// ---------------------------------------------------------------------------
// GCN block (SGConv K=5 + linear + BatchNorm) for MI455X / gfx1250.
// Working set (~23MB) is L2-resident (192MB L2): propagation is bound by
// f32 atomic throughput, not HBM. Final linear uses V_WMMA_F32_16X16X4_F32
// (exact fp32 on the matrix pipe; GEMM cost is trivial so no quantization).
// ---------------------------------------------------------------------------

typedef __attribute__((ext_vector_type(2))) float v2f;
typedef __attribute__((ext_vector_type(8))) float v8f;

#define FDIM   128          // feature dim F == channel dim C
#define BDIM   2            // batch
#define ROWF   (BDIM*FDIM)  // 256 floats per node row in propagation layout
#define KHOPS  5
#define BN_EPS 1e-5f

__device__ __forceinline__ float bcast_f32(float v) {
    return __uint_as_float(__builtin_amdgcn_readfirstlane(__float_as_uint(v)));
}

// --- degree init: self-loop contributes |1.0| at src=n; also clear BN stats
__global__ void k_init(float* __restrict__ deg, float* __restrict__ stats, int N) {
    int i = blockIdx.x * blockDim.x + threadIdx.x;
    if (i < N)   deg[i]   = 1.0f;
    if (i < 256) stats[i] = 0.0f;   // sum[128] + sumsq[128]
}

// --- deg[src] += |w|
__global__ void k_accum_deg(const int* __restrict__ src, const float* __restrict__ w,
                            float* __restrict__ deg, int E) {
    int e = blockIdx.x * blockDim.x + threadIdx.x;
    if (e < E) unsafeAtomicAdd(&deg[src[e]], fabsf(w[e]));
}

// --- dinv = rsqrt(deg), selfnorm = dinv^2 (self-loop edge norm)
__global__ void k_dinv(const float* __restrict__ deg, float* __restrict__ dinv,
                       float* __restrict__ selfn, int N) {
    int i = blockIdx.x * blockDim.x + threadIdx.x;
    if (i >= N) return;
    float d  = deg[i];
    float di = d > 0.0f ? rsqrtf(d) : 0.0f;
    dinv[i]  = di;
    selfn[i] = di * di;
}

// --- norm[e] = dinv[src]*w*dinv[dst] (sign kept in w)
__global__ void k_norm(const int* __restrict__ src, const int* __restrict__ dst,
                       const float* __restrict__ w, const float* __restrict__ dinv,
                       float* __restrict__ norm, int E) {
    int e = blockIdx.x * blockDim.x + threadIdx.x;
    if (e < E) norm[e] = dinv[src[e]] * w[e] * dinv[dst[e]];
}

// --- x[B,N,F] -> h[N, B*F] (node-major: one 1KB contiguous row per node)
__global__ void k_reshape(const float* __restrict__ x, float* __restrict__ h, int N) {
    int t = blockIdx.x * blockDim.x + threadIdx.x;   // N*64 float4 chunks
    if (t >= N * 64) return;
    int d   = t * 4;
    int n   = d >> 8;        // /256
    int rem = d & 255;
    int b   = rem >> 7;
    int f   = rem & 127;
    float4 v = *(const float4*)(x + ((size_t)b * N + n) * FDIM + f);
    *(float4*)(h + d) = v;
}

// --- hop part 1: h_next = selfnorm[n] * h_cur  (also zero-initializes h_next)
// One wave per node; selfnorm broadcast via readfirstlane (wave-uniform).
__global__ __launch_bounds__(256) void k_hop_self(const float* __restrict__ hc,
                                                  float* __restrict__ hn,
                                                  const float* __restrict__ selfn, int N) {
    int wave = blockIdx.x * 8 + (threadIdx.x >> 5);
    if (wave >= N) return;
    int n    = __builtin_amdgcn_readfirstlane(wave);
    int lane = threadIdx.x & 31;
    float s = bcast_f32(selfn[n]);
    const float4* a = (const float4*)(hc + (size_t)n * ROWF);
    float4*       b = (float4*)(hn + (size_t)n * ROWF);
#pragma unroll
    for (int i = 0; i < 2; ++i) {
        int c = lane + i * 32;          // float4 chunk 0..63
        float4 v = a[c];
        v.x *= s; v.y *= s; v.z *= s; v.w *= s;
        b[c] = v;
    }
}

// --- hop part 2: h_next[dst] += norm[e] * h_cur[src].
// One wave per edge: metadata is wave-uniform (scalar broadcast), each lane
// moves two float4 chunks of the 256-float row via native f32 atomics.
__global__ __launch_bounds__(256) void k_hop_edges(const int* __restrict__ src,
                                                   const int* __restrict__ dst,
                                                   const float* __restrict__ norm,
                                                   const float* __restrict__ hc,
                                                   float* __restrict__ hn, int E) {
    int wave = blockIdx.x * 8 + (threadIdx.x >> 5);
    if (wave >= E) return;
    int e    = __builtin_amdgcn_readfirstlane(wave);
    int lane = threadIdx.x & 31;
    int s = __builtin_amdgcn_readfirstlane(src[e]);
    int d = __builtin_amdgcn_readfirstlane(dst[e]);
    float wn = bcast_f32(norm[e]);
    const float4* hs = (const float4*)(hc + (size_t)s * ROWF);
    float*        hd = hn + (size_t)d * ROWF;
#pragma unroll
    for (int i = 0; i < 2; ++i) {
        int c = lane + i * 32;          // float4 chunk 0..63
        float4 v = hs[c];
        float* p = hd + c * 4;
        unsafeAtomicAdd(p + 0, v.x * wn);
        unsafeAtomicAdd(p + 1, v.y * wn);
        unsafeAtomicAdd(p + 2, v.z * wn);
        unsafeAtomicAdd(p + 3, v.w * wn);
    }
}

// --- out[r, c] = sum_k h[r, k] * lin_w[c, k] + lin_b[c], fp32 WMMA 16x16x4.
// One wave per 16x16 output tile. r = b*N + n; tiles never cross b (N%16==0).
__global__ __launch_bounds__(256) void k_gemm_wmma(const float* __restrict__ h,
                                                   const float* __restrict__ lw,
                                                   const float* __restrict__ lb,
                                                   float* __restrict__ out, int N) {
    int lane = threadIdx.x & 31;
    int wv   = threadIdx.x >> 5;
    int tile = blockIdx.x * 8 + wv;
    int totRows = BDIM * N;
    int col0 = (tile & 7) << 4;          // 8 col tiles (C=128)
    int row0 = (tile >> 3) << 4;
    if (row0 >= totRows) return;
    int half = lane >> 4;                // 0: lanes 0-15, 1: lanes 16-31
    int m    = lane & 15;
    int b    = row0 / N;                 // constant over the tile
    int n0   = row0 - b * N;
    const float* arow = h  + (size_t)(n0 + m) * ROWF + b * FDIM; // A row m
    const float* brow = lw + (size_t)(col0 + m) * FDIM;          // B col m (lw row)
    v8f acc = {};
#pragma unroll
    for (int k = 0; k < FDIM; k += 4) {
        // 32-bit A 16x4: lane half selects K pair {0,1} / {2,3}
        v2f a, bb;
        a.x  = arow[k + 2 * half];
        a.y  = arow[k + 2 * half + 1];
        bb.x = brow[k + 2 * half];
        bb.y = brow[k + 2 * half + 1];
        acc = __builtin_amdgcn_wmma_f32_16x16x4_f32(
            /*neg_a=*/false, a, /*neg_b=*/false, bb,
            /*c_mod=*/(short)0, acc, /*reuse_a=*/false, /*reuse_b=*/false);
    }
    float bias = lb[col0 + m];
    // D: VGPR v -> row (row0 + v + 8*half), col (col0 + m)
#pragma unroll
    for (int v = 0; v < 8; ++v) {
        int r = row0 + v + 8 * half;
        out[(size_t)r * FDIM + col0 + m] = acc[v] + bias;
    }
}

// --- BN pass 1: per-channel partial sums over a row range, atomic reduce
__global__ void k_bn_sums(const float* __restrict__ out, float* __restrict__ sum,
                          float* __restrict__ sumsq, int R, int rowsPerBlock) {
    int c  = threadIdx.x;                       // 128 channels
    int r0 = blockIdx.x * rowsPerBlock;
    int r1 = r0 + rowsPerBlock; if (r1 > R) r1 = R;
    float s = 0.0f, s2 = 0.0f;
    for (int r = r0; r < r1; ++r) {
        float v = out[(size_t)r * FDIM + c];
        s += v; s2 += v * v;
    }
    unsafeAtomicAdd(&sum[c],   s);
    unsafeAtomicAdd(&sumsq[c], s2);
}

// --- BN pass 2: fold mean/var + gamma/beta into scale/shift
__global__ void k_bn_finalize(const float* __restrict__ sum, const float* __restrict__ sumsq,
                              const float* __restrict__ gamma, const float* __restrict__ beta,
                              float* __restrict__ scale, float* __restrict__ shift, int R) {
    int c = threadIdx.x;
    float invR = 1.0f / (float)R;
    float mean = sum[c] * invR;
    float var  = sumsq[c] * invR - mean * mean;
    if (var < 0.0f) var = 0.0f;
    float inv = rsqrtf(var + BN_EPS);
    float sc  = gamma[c] * inv;
    scale[c] = sc;
    shift[c] = beta[c] - mean * sc;
}

// --- BN pass 3: out = out*scale[c] + shift[c]
__global__ void k_bn_apply(float* __restrict__ out, const float* __restrict__ scale,
                           const float* __restrict__ shift, int R) {
    int t = blockIdx.x * blockDim.x + threadIdx.x;   // R*32 float4 chunks
    if (t >= R * (FDIM / 4)) return;
    int d = t * 4;
    int c = d & 127;
    float4 v  = *(const float4*)(out + d);
    float4 sc = *(const float4*)(scale + c);
    float4 sh = *(const float4*)(shift + c);
    v.x = v.x * sc.x + sh.x;
    v.y = v.y * sc.y + sh.y;
    v.z = v.z * sc.z + sh.z;
    v.w = v.w * sc.w + sh.w;
    *(float4*)(out + d) = v;
}

extern "C" void kernel_launch(void* const* d_in, const int* in_sizes, int n_in,
                              void* d_out, int out_size, void* d_ws, size_t ws_size,
                              hipStream_t stream) {
    const float* x     = (const float*)d_in[0];
    const int*   edge  = (const int*)d_in[1];   // [2, E]
    const float* w     = (const float*)d_in[2];
    const float* lin_w = (const float*)d_in[3];
    const float* lin_b = (const float*)d_in[4];
    const float* gamma = (const float*)d_in[5];
    const float* beta  = (const float*)d_in[6];

    const int N = in_sizes[0] / (BDIM * FDIM);  // 10000
    const int E = in_sizes[2];                  // 320000
    const int* src = edge;
    const int* dst = edge + E;

    // workspace layout (floats)
    float* ws    = (float*)d_ws;
    float* deg   = ws;                              // N
    float* dinv  = deg   + N;                       // N
    float* selfn = dinv  + N;                       // N
    float* norm  = selfn + N;                       // E
    float* hA    = norm  + E;                       // N*256
    float* hB    = hA    + (size_t)N * ROWF;        // N*256
    float* sum   = hB    + (size_t)N * ROWF;        // 128
    float* sumsq = sum   + FDIM;                    // 128
    float* scale = sumsq + FDIM;                    // 128
    float* shift = scale + FDIM;                    // 128
    float* out   = (float*)d_out;

    const int T = 256;
    k_init<<<(N + T - 1) / T, T, 0, stream>>>(deg, sum, N);
    k_accum_deg<<<(E + T - 1) / T, T, 0, stream>>>(src, w, deg, E);
    k_dinv<<<(N + T - 1) / T, T, 0, stream>>>(deg, dinv, selfn, N);
    k_norm<<<(E + T - 1) / T, T, 0, stream>>>(src, dst, w, dinv, norm, E);
    k_reshape<<<(N * 64 + T - 1) / T, T, 0, stream>>>(x, hA, N);

    float* cur = hA;
    float* nxt = hB;
    for (int hop = 0; hop < KHOPS; ++hop) {
        k_hop_self<<<(N + 7) / 8, T, 0, stream>>>(cur, nxt, selfn, N);      // wave/node
        k_hop_edges<<<(E + 7) / 8, T, 0, stream>>>(src, dst, norm, cur, nxt, E); // wave/edge
        float* tmp = cur; cur = nxt; nxt = tmp;
    }

    const int totRows = BDIM * N;                       // 20000 (divisible by 16)
    const int nTiles  = (totRows / 16) * (FDIM / 16);   // 10000 waves
    k_gemm_wmma<<<(nTiles + 7) / 8, 256, 0, stream>>>(cur, lin_w, lin_b, out, N);

    const int rowsPerBlock = 100;
    k_bn_sums<<<(totRows + rowsPerBlock - 1) / rowsPerBlock, FDIM, 0, stream>>>(
        out, sum, sumsq, totRows, rowsPerBlock);
    k_bn_finalize<<<1, FDIM, 0, stream>>>(sum, sumsq, gamma, beta, scale, shift, totRows);
    k_bn_apply<<<(totRows * (FDIM / 4) + T - 1) / T, T, 0, stream>>>(out, scale, shift, totRows);
}